// SpectralAwareSSM_25082609009154
// MI455X (gfx1250) — compile-verified
//
#include <hip/hip_runtime.h>
#include <hip/hip_bf16.h>
#include <math.h>

#define B_  4
#define L_  2048
#define D_  768
#define N_  16
#define M_  40
#define KX  33          // 2N+1 x_proj outputs
#define KS  17          // N+1 snr outputs
#define NCHUNK 16
#define CLEN   128      // L_/NCHUNK
#define PREC   36       // padded per-(b,l) param record (floats): dtc,eps,floor,pad,Bmod[16],C[16]

typedef __attribute__((ext_vector_type(2))) float v2f;
typedef __attribute__((ext_vector_type(8))) float v8f;

// Workspace layout (bytes)
#define OFF_XP   ((size_t)0)                         // 8192*33*4   = 1,081,344
#define OFF_PAR  ((size_t)1081344)                   // 8192*36*4   = 1,179,648
#define OFF_P    ((size_t)2260992)                   // 4*16*768*16*4 = 3,145,728
#define OFF_S    ((size_t)5406720)
#define OFF_HIN  ((size_t)8552448)                   // end = 11,698,176 bytes

// ---- CDNA5 async global->LDS copy (ASYNCcnt), guarded so compile never breaks
#if defined(__has_builtin)
#if __has_builtin(__builtin_amdgcn_global_load_async_to_lds_b128) && \
    __has_builtin(__builtin_amdgcn_s_wait_asynccnt)
#define HAVE_ASYNC_LDS 1
#endif
#endif
#ifndef HAVE_ASYNC_LDS
#define HAVE_ASYNC_LDS 0
#endif

#if HAVE_ASYNC_LDS
// Builtin signature (from hipcc diagnostic): param 1 is
// 'int __attribute__((vector_size(16))) __device__ *' i.e. addrspace(1) v4i*.
typedef int v4i __attribute__((vector_size(16)));
typedef __attribute__((address_space(1))) v4i* gas_ptr;
typedef __attribute__((address_space(3))) v4i* las_ptr;
#endif

// Stage CLEN*PREC floats of params into LDS (16B-aligned both sides).
__device__ __forceinline__ void stage_params(float* sp, const float* pb, int tid) {
#if HAVE_ASYNC_LDS
  for (int i = tid; i < (CLEN * PREC) / 4; i += 256) {
    __builtin_amdgcn_global_load_async_to_lds_b128(
        (gas_ptr)(pb + 4 * i), (las_ptr)(sp + 4 * i), 0, 0);
  }
  __builtin_amdgcn_s_wait_asynccnt(0);
#else
  for (int i = tid; i < CLEN * PREC; i += 256) sp[i] = pb[i];
#endif
  __syncthreads();
}

// ---------------------------------------------------------------------------
// Kernel A: x_proj[row][k] = sum_d x[row][d] * Wx[k][d]  via WMMA f32 16x16x4.
// Wave = one 16-row slab, 3 column tiles (k padded 33->48, zeros in LDS).
// A fragment layout (ISA 16x4 f32): lanes 0-15 hold K=0,1; lanes 16-31 K=2,3.
// LDS holds W^T as paired K-values so each B fragment is ONE ds_load_b64
// into an even-aligned VGPR pair (no shuffle movs before v_wmma).
// ---------------------------------------------------------------------------
__global__ __launch_bounds__(256) void kA_xproj(const float* __restrict__ x,
                                                const float* __restrict__ Wx,
                                                float* __restrict__ xp) {
  __shared__ float2 sW2[64 * 48];          // [dpair][k]: (Wx[k][d], Wx[k][d+1]), k padded to 48
  const int tid  = threadIdx.x;
  const int lane = tid & 31;
  const int wave = tid >> 5;
  const int hb   = lane >> 4;              // half-wave select
  const int l16  = lane & 15;
  const int rowBase = blockIdx.x * 128 + wave * 16;
  const float2* Wx2 = (const float2*)Wx;   // Wx rows are 768 floats (even, 8B-aligned)

  v8f acc0 = {0.f,0.f,0.f,0.f,0.f,0.f,0.f,0.f};
  v8f acc1 = acc0, acc2 = acc0;
  const float2 z2 = {0.f, 0.f};

  for (int kc = 0; kc < 6; ++kc) {         // 6 chunks of 128 over K=768
    const int dc  = kc * 128;
    const int dc2 = kc * 64;
    for (int i = tid; i < 64 * 48; i += 256) {
      int dp = i / 48, k = i % 48;
      sW2[i] = (k < KX) ? Wx2[(size_t)k * 384 + dc2 + dp] : z2;
    }
    __syncthreads();
    for (int d0 = 0; d0 < 128; d0 += 4) {
      const int dp0 = (d0 >> 1) + hb;      // dpair index for this lane's K pair
      const v2f a  = *(const v2f*)(x + (size_t)(rowBase + l16) * D_ + dc + d0 + (hb << 1));
      const v2f b0 = *(const v2f*)&sW2[dp0 * 48 + l16];
      const v2f b1 = *(const v2f*)&sW2[dp0 * 48 + 16 + l16];
      const v2f b2 = *(const v2f*)&sW2[dp0 * 48 + 32 + l16];
      acc0 = __builtin_amdgcn_wmma_f32_16x16x4_f32(false, a, false, b0, (short)0, acc0, false, false);
      acc1 = __builtin_amdgcn_wmma_f32_16x16x4_f32(false, a, false, b1, (short)0, acc1, false, false);
      acc2 = __builtin_amdgcn_wmma_f32_16x16x4_f32(false, a, false, b2, (short)0, acc2, false, false);
    }
    __syncthreads();
  }
  // C/D layout: vgpr r holds row (r + 8*halfwave), col = lane&15
  for (int r = 0; r < 8; ++r) {
    const int m = rowBase + r + (hb << 3);
    xp[(size_t)m * KX + l16]      = acc0[r];
    xp[(size_t)m * KX + 16 + l16] = acc1[r];
    if (32 + l16 < KX) xp[(size_t)m * KX + 32 + l16] = acc2[r];
  }
}

// ---------------------------------------------------------------------------
// Kernel B: per-(b,l) parameter records from snr_mel + x_proj.
// ---------------------------------------------------------------------------
__global__ __launch_bounds__(256) void kB_params(const float* __restrict__ snr,
                                                 const float* __restrict__ Wsnr,
                                                 const float* __restrict__ bsnr,
                                                 const float* __restrict__ xp,
                                                 const float* __restrict__ alphap,
                                                 float* __restrict__ params) {
  __shared__ float sWs[KS * M_];
  __shared__ float sb[KS];
  const int tid = threadIdx.x;
  for (int i = tid; i < KS * M_; i += 256) sWs[i] = Wsnr[i];
  if (tid < KS) sb[tid] = bsnr[tid];
  __syncthreads();

  const int row = blockIdx.x * 256 + tid;   // < 8192
  float s[M_];
  float mean = 0.f;
  const float* sr = snr + (size_t)row * M_;
#pragma unroll
  for (int m = 0; m < M_; ++m) { s[m] = sr[m]; mean += s[m]; }
  mean *= (1.f / (float)M_);
  const float alpha = alphap[0];

  float sm[KS];
  for (int k = 0; k < KS; ++k) {
    float acc = sb[k];
#pragma unroll
    for (int m = 0; m < M_; ++m) acc = fmaf(s[m], sWs[k * M_ + m], acc);
    sm[k] = acc;
  }

  const float* xr = xp + (size_t)row * KX;
  float* rec = params + (size_t)row * PREC;
  rec[0] = xr[0] + sm[0];                              // dtc = dt_raw + dt_snr_shift
  rec[1] = 0.2f  - (0.2f  - 0.08f) * mean;             // adaptive_eps
  rec[2] = 0.05f + (0.15f - 0.05f) * mean;             // adaptive_floor
  rec[3] = 0.f;
#pragma unroll
  for (int n = 0; n < N_; ++n) {
    float g    = 1.f / (1.f + __expf(-sm[1 + n]));     // sigmoid
    float gate = g * 0.7f + 0.3f;                      // *(1-BGATE_FLOOR)+BGATE_FLOOR
    rec[4 + n]  = xr[1 + n] * (1.f - alpha + alpha * gate);   // Bmod
    rec[20 + n] = xr[17 + n];                                 // C
  }
}

__device__ __forceinline__ float softplusf(float z) {
  float e = __expf(z);
  return (z > 20.f) ? z : __logf(1.f + e);
}

// ---------------------------------------------------------------------------
// Kernel C: pass 1 — per-chunk local scan with h0=0; emit P=exp(A*sum(delta))
// and s = final local state.  grid = (3 d-tiles, 16 chunks, 4 batches).
// ---------------------------------------------------------------------------
__global__ __launch_bounds__(256) void kC_scan1(const float* __restrict__ x,
                                                const float* __restrict__ params,
                                                const float* __restrict__ Wdt,
                                                const float* __restrict__ bdt,
                                                const float* __restrict__ A_log,
                                                float* __restrict__ P,
                                                float* __restrict__ S) {
  __shared__ __align__(16) float sp[CLEN * PREC];      // 18 KB param table
  const int tid = threadIdx.x;
  const int b = blockIdx.z, c = blockIdx.y;
  const int d = blockIdx.x * 256 + tid;
  const int t0 = c * CLEN;
  stage_params(sp, params + ((size_t)b * L_ + t0) * PREC, tid);

  const float wdt = Wdt[d], bd = bdt[d];
  float A[N_], h[N_];
#pragma unroll
  for (int n = 0; n < N_; ++n) { A[n] = -__expf(A_log[(size_t)d * N_ + n]); h[n] = 0.f; }

  float dsum = 0.f;
  const float* xb = x + ((size_t)b * L_ + t0) * D_ + d;
  for (int tt = 0; tt < CLEN; ++tt) {
    float xv = xb[(size_t)tt * D_];
    int tn = (tt + 2 < CLEN) ? tt + 2 : CLEN - 1;
    __builtin_prefetch(xb + (size_t)tn * D_, 0, 0);
    const float* p = sp + tt * PREC;
    float delta = softplusf(fmaf(p[0], wdt, bd)) + p[2];
    float eps = p[1];
    dsum += delta;
#pragma unroll
    for (int n = 0; n < N_; ++n) {
      float dA   = __expf(A[n] * delta);
      float coef = fmaf(delta, p[4 + n], eps);
      h[n] = fmaf(dA, h[n], coef * xv);
    }
  }
  const size_t base = (((size_t)b * NCHUNK + c) * D_ + d) * N_;
#pragma unroll
  for (int n = 0; n < N_; ++n) {
    P[base + n] = __expf(A[n] * dsum);
    S[base + n] = h[n];
  }
}

// ---------------------------------------------------------------------------
// Kernel D: serial combine over 16 chunks; Hin[c] = state entering chunk c.
// One thread per (b,d,n): 49152 threads.
// ---------------------------------------------------------------------------
__global__ __launch_bounds__(256) void kD_combine(const float* __restrict__ P,
                                                  const float* __restrict__ S,
                                                  float* __restrict__ Hin) {
  const int idx = blockIdx.x * 256 + threadIdx.x;    // < B_*D_*N_
  const int b = idx / (D_ * N_);
  const int r = idx % (D_ * N_);
  float h = 0.f;
  for (int c = 0; c < NCHUNK; ++c) {
    const size_t o = ((size_t)b * NCHUNK + c) * (size_t)(D_ * N_) + r;
    Hin[o] = h;
    h = fmaf(P[o], h, S[o]);
  }
}

// ---------------------------------------------------------------------------
// Kernel E: pass 2 — rerun each chunk with correct incoming state; emit y.
// ---------------------------------------------------------------------------
__global__ __launch_bounds__(256) void kE_scan2(const float* __restrict__ x,
                                                const float* __restrict__ params,
                                                const float* __restrict__ Wdt,
                                                const float* __restrict__ bdt,
                                                const float* __restrict__ A_log,
                                                const float* __restrict__ Dp,
                                                const float* __restrict__ Hin,
                                                float* __restrict__ out) {
  __shared__ __align__(16) float sp[CLEN * PREC];
  const int tid = threadIdx.x;
  const int b = blockIdx.z, c = blockIdx.y;
  const int d = blockIdx.x * 256 + tid;
  const int t0 = c * CLEN;
  stage_params(sp, params + ((size_t)b * L_ + t0) * PREC, tid);

  const float wdt = Wdt[d], bd = bdt[d], dp = Dp[d];
  const size_t base = (((size_t)b * NCHUNK + c) * D_ + d) * N_;
  float A[N_], h[N_];
#pragma unroll
  for (int n = 0; n < N_; ++n) {
    A[n] = -__expf(A_log[(size_t)d * N_ + n]);
    h[n] = Hin[base + n];
  }

  const float* xb = x + ((size_t)b * L_ + t0) * D_ + d;
  float* yb = out + ((size_t)b * L_ + t0) * D_ + d;
  for (int tt = 0; tt < CLEN; ++tt) {
    float xv = xb[(size_t)tt * D_];
    int tn = (tt + 2 < CLEN) ? tt + 2 : CLEN - 1;
    __builtin_prefetch(xb + (size_t)tn * D_, 0, 0);
    const float* p = sp + tt * PREC;
    float delta = softplusf(fmaf(p[0], wdt, bd)) + p[2];
    float eps = p[1];
    float acc = dp * xv;
#pragma unroll
    for (int n = 0; n < N_; ++n) {
      float dA   = __expf(A[n] * delta);
      float coef = fmaf(delta, p[4 + n], eps);
      h[n] = fmaf(dA, h[n], coef * xv);
      acc  = fmaf(h[n], p[20 + n], acc);
    }
    yb[(size_t)tt * D_] = acc;
  }
}

// ---------------------------------------------------------------------------
extern "C" void kernel_launch(void* const* d_in, const int* in_sizes, int n_in,
                              void* d_out, int out_size, void* d_ws, size_t ws_size,
                              hipStream_t stream) {
  const float* x     = (const float*)d_in[0];
  const float* snr   = (const float*)d_in[1];
  const float* Wx    = (const float*)d_in[2];
  const float* Wsnr  = (const float*)d_in[3];
  const float* bsnr  = (const float*)d_in[4];
  const float* Wdt   = (const float*)d_in[5];
  const float* bdt   = (const float*)d_in[6];
  const float* A_log = (const float*)d_in[7];
  const float* Dp    = (const float*)d_in[8];
  const float* alpha = (const float*)d_in[9];
  float* out = (float*)d_out;

  char* ws = (char*)d_ws;
  float* xp     = (float*)(ws + OFF_XP);
  float* params = (float*)(ws + OFF_PAR);
  float* P      = (float*)(ws + OFF_P);
  float* S      = (float*)(ws + OFF_S);
  float* Hin    = (float*)(ws + OFF_HIN);

  kA_xproj  <<<dim3(64),            dim3(256), 0, stream>>>(x, Wx, xp);
  kB_params <<<dim3(32),            dim3(256), 0, stream>>>(snr, Wsnr, bsnr, xp, alpha, params);
  kC_scan1  <<<dim3(3, NCHUNK, B_), dim3(256), 0, stream>>>(x, params, Wdt, bdt, A_log, P, S);
  kD_combine<<<dim3(192),           dim3(256), 0, stream>>>(P, S, Hin);
  kE_scan2  <<<dim3(3, NCHUNK, B_), dim3(256), 0, stream>>>(x, params, Wdt, bdt, A_log, Dp, Hin, out);
}